// GNN_44702019617183
// MI455X (gfx1250) — compile-verified
//
#include <hip/hip_runtime.h>

// ---------------------------------------------------------------------------
// GraphConv GNN for MI455X (gfx1250): fp32 WMMA 16x16x4 GEMM + L2 atomic SpMM
// ---------------------------------------------------------------------------

#define N_NODES 50000
#define N_EDGES 800000
#define N_GRAPHS 64
#define F_IN 128
#define F_HID 96
#define KP_STRIDE 112   // float2 stride per k-pair row: 112*8B -> +32 bank shift
                        // between kp and kp+1 -> conflict-free halves (64 banks)

typedef __attribute__((ext_vector_type(2))) float v2f;
typedef __attribute__((ext_vector_type(8))) float v8f;

// ---------------- zero scratch ----------------
__global__ void gnn_zero_kernel(float* __restrict__ p, int n) {
    int i = blockIdx.x * blockDim.x + threadIdx.x;
    if (i < n) p[i] = 0.0f;
}

// ---------------- degree accumulation (scatter-add of ones) ----------------
__global__ void gnn_degree_kernel(const int* __restrict__ src, const int* __restrict__ dst,
                                  float* __restrict__ degS, float* __restrict__ degD, int E) {
    int e = blockIdx.x * blockDim.x + threadIdx.x;
    if (e < E) {
        atomicAdd(degS + src[e], 1.0f);
        atomicAdd(degD + dst[e], 1.0f);
    }
}

// ---------------- deg -> clip(deg,1)^-0.5 in place (both arrays contiguous) --
__global__ void gnn_norm_kernel(float* __restrict__ deg, int n) {
    int i = blockIdx.x * blockDim.x + threadIdx.x;
    if (i < n) {
        float v = deg[i];
        v = v < 1.0f ? 1.0f : v;
        deg[i] = rsqrtf(v);
    }
}

// ---------------- GEMM: H[N,96] = (X[N,K] * norm_src[:,None]) @ W[K,96] ------
// One wave computes a 16(M) x 96(N) stripe: 6 WMMA C accumulators, A fragment
// reused across all 6 N-tiles. W staged in LDS packed as k-pair float2s so
// every B fragment is a single aligned ds_load_b64 (no VGPR shuffles).
// A is software-pipelined one k-step ahead to hide global load latency.
template <int K>
__global__ __launch_bounds__(256) void gnn_gemm_wmma_kernel(
        const float* __restrict__ X, const float* __restrict__ nrm,
        const float* __restrict__ W, float* __restrict__ Hout, int mtiles) {
    // (K_max/2) kp-rows * padded stride * 8B = 64*112*8 = 57344 B (< 320KB/WGP)
    __shared__ v2f sW[(F_IN / 2) * KP_STRIDE];

    const int tid = threadIdx.x;
    // cooperative load + pack: sW[kp*STR + n] = {W[2kp][n], W[2kp+1][n]}
    for (int i = tid; i < (K / 2) * F_HID; i += 256) {
        int kp = i / F_HID;
        int n  = i - kp * F_HID;
        v2f w;
        w.x = W[(2 * kp) * F_HID + n];
        w.y = W[(2 * kp + 1) * F_HID + n];
        sW[kp * KP_STRIDE + n] = w;
    }
    __syncthreads();

    const int wave  = tid >> 5;
    const int lane  = tid & 31;
    const int mtile = blockIdx.x * 8 + wave;
    if (mtile >= mtiles) return;  // no barriers past this point

    const int m0   = mtile << 4;
    const int mrow = m0 + (lane & 15);         // A row this lane supplies
    const int kh   = lane >> 4;                // 0: K={k0,k0+1}, 1: K={k0+2,k0+3}
    const float ns = nrm[mrow];                // row scale (out-degree^-1/2)

    // A: X row as float2 stream (8B aligned: K*4 and kh*8 both multiples of 8)
    const v2f* __restrict__ xv = (const v2f*)(X + mrow * K) + kh;
    // B: col base in packed-W LDS image
    const v2f* __restrict__ wb = sW + kh * KP_STRIDE + (lane & 15);

    v8f c[6];
#pragma unroll
    for (int t = 0; t < 6; ++t) c[t] = (v8f){0.f,0.f,0.f,0.f,0.f,0.f,0.f,0.f};

    v2f areg = xv[0];                          // pipeline prologue
    for (int k0 = 0; k0 < K; k0 += 4) {
        v2f a = areg * ns;                     // scaled A fragment (16x4 fp32)
        if (k0 + 4 < K) areg = xv[(k0 >> 1) + 2];  // prefetch next k-step
        const v2f* __restrict__ wk = wb + (k0 >> 1) * KP_STRIDE;
#pragma unroll
        for (int t = 0; t < 6; ++t) {
            v2f b = wk[t * 16];                // single ds_load_b64
            c[t] = __builtin_amdgcn_wmma_f32_16x16x4_f32(
                       false, a, false, b, (short)0, c[t], false, false);
        }
    }

    // C/D layout: VGPR v -> row m0 + v + 8*(lane>>4), col = lane&15 within tile
    const int obase = (m0 + (kh << 3)) * F_HID + (lane & 15);
#pragma unroll
    for (int v = 0; v < 8; ++v) {
        float* __restrict__ o = Hout + obase + v * F_HID;
        o[0]  = c[0][v];
        o[16] = c[1][v];
        o[32] = c[2][v];
        o[48] = c[3][v];
        o[64] = c[4][v];
        o[80] = c[5][v];
    }
}

// ---------------- SpMM aggregate: Agg[dst] += H[src]  (one wave per edge) ----
__global__ void gnn_spmm_kernel(const int* __restrict__ src, const int* __restrict__ dst,
                                const float* __restrict__ Hin, float* __restrict__ Agg,
                                int E) {
    int gid  = blockIdx.x * blockDim.x + threadIdx.x;
    int e    = gid >> 5;
    int lane = gid & 31;
    if (e >= E) return;
    int s = src[e] * F_HID;
    int d = dst[e] * F_HID;
    // 96 cols = 3 per lane; each instruction covers a contiguous 128B segment
    // across the wave; atomics resolve in L2 (feature array is L2-resident)
    atomicAdd(&Agg[d + lane],      Hin[s + lane]);
    atomicAdd(&Agg[d + lane + 32], Hin[s + lane + 32]);
    atomicAdd(&Agg[d + lane + 64], Hin[s + lane + 64]);
}

// ---------------- x = relu(agg * norm_dst + b) in place ----------------------
__global__ void gnn_bias_relu_kernel(float* __restrict__ A, const float* __restrict__ nrmD,
                                     const float* __restrict__ bias, int total) {
    int i = blockIdx.x * blockDim.x + threadIdx.x;
    if (i >= total) return;
    int n = i / F_HID;
    int cidx = i - n * F_HID;
    float v = A[i] * nrmD[n] + bias[cidx];
    A[i] = v > 0.0f ? v : 0.0f;
}

// ---------------- per-graph sums + counts ------------------------------------
__global__ void gnn_readout_kernel(const float* __restrict__ X, const int* __restrict__ gids,
                                   float* __restrict__ gsum, float* __restrict__ gcnt,
                                   int total) {
    int i = blockIdx.x * blockDim.x + threadIdx.x;
    if (i >= total) return;
    int n = i / F_HID;
    int cidx = i - n * F_HID;
    int g = gids[n];
    atomicAdd(&gsum[g * F_HID + cidx], X[i]);
    if (cidx == 0) atomicAdd(&gcnt[g], 1.0f);
}

__global__ void gnn_finalize_kernel(const float* __restrict__ gsum,
                                    const float* __restrict__ gcnt,
                                    float* __restrict__ out) {
    int i = blockIdx.x * blockDim.x + threadIdx.x;
    if (i >= N_GRAPHS * F_HID) return;
    int g = i / F_HID;
    float c = gcnt[g];
    c = c < 1.0f ? 1.0f : c;
    out[i] = gsum[i] / c;
}

// ---------------------------------------------------------------------------
extern "C" void kernel_launch(void* const* d_in, const int* in_sizes, int n_in,
                              void* d_out, int out_size, void* d_ws, size_t ws_size,
                              hipStream_t stream) {
    const float* feat = (const float*)d_in[0];   // [50000,128]
    const float* W1   = (const float*)d_in[1];   // [128,96]
    const float* b1   = (const float*)d_in[2];   // [96]
    const float* W2   = (const float*)d_in[3];   // [96,96]
    const float* b2   = (const float*)d_in[4];   // [96]
    const int*   src  = (const int*)d_in[5];     // [800000]
    const int*   dst  = (const int*)d_in[6];     // [800000]
    const int*   gids = (const int*)d_in[7];     // [50000]

    // workspace layout
    float* bufA = (float*)d_ws;                      // N*96  (h / h2)
    float* bufB = bufA + N_NODES * F_HID;            // N*96  (agg / x)
    float* degS = bufB + N_NODES * F_HID;            // N     (-> norm_src)
    float* degD = degS + N_NODES;                    // N     (-> norm_dst)
    float* gsum = degD + N_NODES;                    // G*96
    float* gcnt = gsum + N_GRAPHS * F_HID;           // G

    const int NH     = N_NODES * F_HID;
    const int mtiles = N_NODES / 16;                 // 3125 (exact)
    const int gemmBlocks = (mtiles + 7) / 8;         // 8 waves/block

    // 1) zero everything atomics accumulate into: bufB .. gcnt (contiguous)
    int ztail = NH + 2 * N_NODES + N_GRAPHS * F_HID + N_GRAPHS;
    gnn_zero_kernel<<<(ztail + 255) / 256, 256, 0, stream>>>(bufB, ztail);

    // 2) degrees + norms
    gnn_degree_kernel<<<(N_EDGES + 255) / 256, 256, 0, stream>>>(src, dst, degS, degD, N_EDGES);
    gnn_norm_kernel<<<(2 * N_NODES + 255) / 256, 256, 0, stream>>>(degS, 2 * N_NODES);

    // 3) layer 1: GEMM (K=128) -> SpMM -> bias/norm/relu
    gnn_gemm_wmma_kernel<F_IN><<<gemmBlocks, 256, 0, stream>>>(feat, degS, W1, bufA, mtiles);
    gnn_spmm_kernel<<<(N_EDGES * 32 + 255) / 256, 256, 0, stream>>>(src, dst, bufA, bufB, N_EDGES);
    gnn_bias_relu_kernel<<<(NH + 255) / 256, 256, 0, stream>>>(bufB, degD, b1, NH);

    // 4) layer 2: GEMM (K=96) -> zero agg -> SpMM -> bias/norm/relu
    gnn_gemm_wmma_kernel<F_HID><<<gemmBlocks, 256, 0, stream>>>(bufB, degS, W2, bufA, mtiles);
    gnn_zero_kernel<<<(NH + 255) / 256, 256, 0, stream>>>(bufB, NH);
    gnn_spmm_kernel<<<(N_EDGES * 32 + 255) / 256, 256, 0, stream>>>(src, dst, bufA, bufB, N_EDGES);
    gnn_bias_relu_kernel<<<(NH + 255) / 256, 256, 0, stream>>>(bufB, degD, b2, NH);

    // 5) per-graph mean readout
    gnn_readout_kernel<<<(NH + 255) / 256, 256, 0, stream>>>(bufB, gids, gsum, gcnt, NH);
    gnn_finalize_kernel<<<(N_GRAPHS * F_HID + 255) / 256, 256, 0, stream>>>(gsum, gcnt, (float*)d_out);
}